// MixerLayer_20779051778729
// MI455X (gfx1250) — compile-verified
//
#include <hip/hip_runtime.h>

typedef float v2f __attribute__((ext_vector_type(2)));
typedef float v8f __attribute__((ext_vector_type(8)));

#define BQ  16
#define VQ  512
#define SQ  1024
#define EQ  8
#define FFQ 2048
#define TOK (BQ * VQ)   // 8192 tokens for MLPTime
#define M2  (BQ * SQ)   // 16384 rows for MLPFeat
#define LDK 36          // [row][k] tile stride (dwords): 16B-aligned, conflict-free
#define LDX 136         // [k][token] X-tile stride (dwords): 16B-aligned, conflict-free

#if defined(__has_builtin)
#if __has_builtin(__builtin_amdgcn_global_load_async_to_lds_b128)
#define USE_ASYNC 1
#endif
#if __has_builtin(__builtin_amdgcn_s_wait_asynccnt)
#define HAVE_WAIT_ASYNC 1
#endif
#endif

// Matches the builtin's parameter pointee type exactly (from hipcc diagnostic):
// __attribute__((__vector_size__(4 * sizeof(int)))) int
typedef int v4i_vs __attribute__((__vector_size__(4 * sizeof(int))));

// 16B global -> LDS copy: async CDNA5 path if available, register path otherwise.
__device__ __forceinline__ void g2l_b128(void* lds, const void* g) {
#ifdef USE_ASYNC
    __builtin_amdgcn_global_load_async_to_lds_b128(
        (__attribute__((address_space(1))) v4i_vs*)g,
        (__attribute__((address_space(3))) v4i_vs*)lds, 0, 0);
#else
    *(float4*)lds = *(const float4*)g;
#endif
}

__device__ __forceinline__ void async_join() {
#ifdef USE_ASYNC
#ifdef HAVE_WAIT_ASYNC
    __builtin_amdgcn_s_wait_asynccnt(0);
#else
    asm volatile("s_wait_asynccnt 0" ::: "memory");
#endif
#endif
}

__device__ __forceinline__ v8f wmma4(v2f a, v2f b, v8f c) {
    // D = A(16x4,f32) * B(4x16,f32) + C(16x16,f32)
    return __builtin_amdgcn_wmma_f32_16x16x4_f32(false, a, false, b, (short)0, c,
                                                 false, false);
}

__device__ __forceinline__ v8f v8zero() {
    v8f z = {0.f, 0.f, 0.f, 0.f, 0.f, 0.f, 0.f, 0.f};
    return z;
}

// ---------------------------------------------------------------------------
// Kernel 1: gating GEMM (WMMA, experts padded to 16 cols) + softmax + top-2
// logits[t,e] = sum_s x[b,s,v] * gate_W[e,s], t = b*V + v
// ---------------------------------------------------------------------------
__global__ __launch_bounds__(256) void gate_softmax_topk(
    const float* __restrict__ x, const float* __restrict__ gW,
    float* __restrict__ probs, float* __restrict__ fullw) {
    __shared__ float Xs[32 * LDX];   // [k][token]
    __shared__ float Gs[16 * LDK];   // [expert][k], rows 8..15 zero
    __shared__ float Ls[128 * 16];

    const int tid = threadIdx.x;
    const int tokBase = blockIdx.x * 128;
    const int b = tokBase >> 9;
    const int v0 = tokBase & 511;
    const int wid = tid >> 5, lane = tid & 31;
    const int lr = lane & 15;
    const int hk = (lane >> 4) * 2;
    const int mtB = wid * 16;

    // zero pad rows 8..15 of Gs once
    {
        int e = 8 + (tid >> 5), sl = tid & 31;
        Gs[e * LDK + sl] = 0.f;
    }

    v8f acc = v8zero();

    for (int kc = 0; kc < SQ; kc += 32) {
        // X tile: 32 k-rows x 128 tokens (token-contiguous b128 loads)
#pragma unroll
        for (int p = 0; p < 4; ++p) {
            int kl = p * 8 + (tid >> 5);
            int t4 = (tid & 31) * 4;
            g2l_b128(&Xs[kl * LDX + t4],
                     &x[(size_t)(b * SQ + kc + kl) * VQ + v0 + t4]);
        }
        // gate_W tile: 8 real rows x 32 k
        if (tid < 64) {
            int e = tid >> 3, s4 = (tid & 7) * 4;
            g2l_b128(&Gs[e * LDK + s4], &gW[(size_t)e * SQ + kc + s4]);
        }
        async_join();
        __syncthreads();
#pragma unroll
        for (int kk = 0; kk < 32; kk += 4) {
            v2f a, bb;
            a.x = Xs[(kk + hk) * LDX + mtB + lr];
            a.y = Xs[(kk + hk + 1) * LDX + mtB + lr];
            bb.x = Gs[lr * LDK + kk + hk];
            bb.y = Gs[lr * LDK + kk + hk + 1];
            acc = wmma4(a, bb, acc);
        }
        __syncthreads();
    }
    // spill logits to LDS
#pragma unroll
    for (int i = 0; i < 8; ++i) {
        int m = mtB + (lane >> 4) * 8 + i;
        Ls[m * 16 + lr] = acc[i];
    }
    __syncthreads();

    if (tid < 128) {
        float p[EQ];
        float mx = -3.4e38f;
#pragma unroll
        for (int e = 0; e < EQ; ++e) {
            p[e] = Ls[tid * 16 + e];
            mx = fmaxf(mx, p[e]);
        }
        float sum = 0.f;
#pragma unroll
        for (int e = 0; e < EQ; ++e) {
            p[e] = __expf(p[e] - mx);
            sum += p[e];
        }
        float inv = 1.f / sum;
#pragma unroll
        for (int e = 0; e < EQ; ++e) p[e] *= inv;
        // top-2, first index wins ties (matches lax.top_k)
        int i0 = 0;
        float b0 = p[0];
#pragma unroll
        for (int e = 1; e < EQ; ++e)
            if (p[e] > b0) { b0 = p[e]; i0 = e; }
        int i1 = -1;
        float b1 = -1.f;
#pragma unroll
        for (int e = 0; e < EQ; ++e)
            if (e != i0 && p[e] > b1) { b1 = p[e]; i1 = e; }

        const int t = tokBase + tid;
#pragma unroll
        for (int e = 0; e < EQ; ++e) {
            probs[t * EQ + e] = p[e];
            fullw[t * EQ + e] = (e == i0) ? b0 : ((e == i1) ? b1 : 0.f);
        }
    }
}

// ---------------------------------------------------------------------------
// Kernel 2: MoE — dense all-expert GEMM with per-row top-k weight fold.
// moe[t,o] = relu( sum_e w[t,e] * ( sum_s xt[t,s]*exp_W[e,o,s] + exp_b[e,o] ) )
// Block: 128 tokens x 128 out dims, 8 waves each 64x32 (4x2 WMMA tiles).
// ---------------------------------------------------------------------------
__global__ __launch_bounds__(256) void moe_expert_gemm(
    const float* __restrict__ x, const float* __restrict__ expW,
    const float* __restrict__ expB, const float* __restrict__ fullw,
    float* __restrict__ moe) {
    __shared__ float Xs[32 * LDX];    // [k][token]
    __shared__ float Ws[128 * LDK];   // [out][k]
    __shared__ float Wt[128 * EQ];

    const int tid = threadIdx.x;
    const int tokBase = blockIdx.x * 128;
    const int outBase = blockIdx.y * 128;
    const int b = tokBase >> 9;
    const int v0 = tokBase & 511;
    const int wid = tid >> 5, lane = tid & 31;
    const int lr = lane & 15;
    const int hk = (lane >> 4) * 2;
    const int wr = (wid >> 2) * 64;  // wave token offset within block
    const int wc = (wid & 3) * 32;   // wave out offset within block

    // per-token dense expert weights (128*8 floats)
    {
        int t4 = tid * 4;
        *(float4*)&Wt[t4] = *(const float4*)&fullw[(size_t)tokBase * EQ + t4];
    }

    v8f acc2[4][2];
#pragma unroll
    for (int mi = 0; mi < 4; ++mi)
#pragma unroll
        for (int ni = 0; ni < 2; ++ni) acc2[mi][ni] = v8zero();

    for (int e = 0; e < EQ; ++e) {
        v8f accE[4][2];
#pragma unroll
        for (int mi = 0; mi < 4; ++mi)
#pragma unroll
            for (int ni = 0; ni < 2; ++ni) accE[mi][ni] = v8zero();

        for (int kc = 0; kc < SQ; kc += 32) {
            // X tile (implicit transpose of x), token-contiguous b128
#pragma unroll
            for (int p = 0; p < 4; ++p) {
                int kl = p * 8 + (tid >> 5);
                int t4 = (tid & 31) * 4;
                g2l_b128(&Xs[kl * LDX + t4],
                         &x[(size_t)(b * SQ + kc + kl) * VQ + v0 + t4]);
            }
            // expert weight tile [128 out x 32 k], k-contiguous b128
#pragma unroll
            for (int p = 0; p < 4; ++p) {
                int lin = p * 256 + tid;
                int ol = lin >> 3, s4 = (lin & 7) * 4;
                g2l_b128(&Ws[ol * LDK + s4],
                         &expW[(size_t)(e * SQ + outBase + ol) * SQ + kc + s4]);
            }
            async_join();
            __syncthreads();
#pragma unroll
            for (int kk = 0; kk < 32; kk += 4) {
                v2f a[4], bb[2];
#pragma unroll
                for (int mi = 0; mi < 4; ++mi) {
                    a[mi].x = Xs[(kk + hk) * LDX + wr + mi * 16 + lr];
                    a[mi].y = Xs[(kk + hk + 1) * LDX + wr + mi * 16 + lr];
                }
#pragma unroll
                for (int ni = 0; ni < 2; ++ni) {
                    bb[ni].x = Ws[(wc + ni * 16 + lr) * LDK + kk + hk];
                    bb[ni].y = Ws[(wc + ni * 16 + lr) * LDK + kk + hk + 1];
                }
#pragma unroll
                for (int mi = 0; mi < 4; ++mi)
#pragma unroll
                    for (int ni = 0; ni < 2; ++ni)
                        accE[mi][ni] = wmma4(a[mi], bb[ni], accE[mi][ni]);
            }
            __syncthreads();
        }
        // fold expert result into persistent accumulator with per-row weight
#pragma unroll
        for (int ni = 0; ni < 2; ++ni) {
            float be = expB[e * SQ + outBase + wc + ni * 16 + lr];
#pragma unroll
            for (int mi = 0; mi < 4; ++mi) {
#pragma unroll
                for (int i = 0; i < 8; ++i) {
                    int ml = wr + mi * 16 + (lane >> 4) * 8 + i;
                    float w = Wt[ml * EQ + e];
                    acc2[mi][ni][i] += w * (accE[mi][ni][i] + be);
                }
            }
        }
    }
    // ReLU epilogue, store [token][out]
#pragma unroll
    for (int mi = 0; mi < 4; ++mi)
#pragma unroll
        for (int ni = 0; ni < 2; ++ni)
#pragma unroll
            for (int i = 0; i < 8; ++i) {
                int m = tokBase + wr + mi * 16 + (lane >> 4) * 8 + i;
                int n = outBase + wc + ni * 16 + lr;
                moe[(size_t)m * SQ + n] = fmaxf(acc2[mi][ni][i], 0.f);
            }
}

// ---------------------------------------------------------------------------
// Kernel 3: transpose [B,V,S] -> [B,S,V] + residual add
// ---------------------------------------------------------------------------
__global__ __launch_bounds__(256) void transpose_res(
    const float* __restrict__ moe, const float* __restrict__ x,
    float* __restrict__ x2) {
    __shared__ float T[32][33];
    const int s0 = blockIdx.x * 32, v0 = blockIdx.y * 32, b = blockIdx.z;
    const int tx = threadIdx.x & 31, ty = threadIdx.x >> 5;
#pragma unroll
    for (int r = 0; r < 4; ++r) {
        int i = r * 8 + ty;
        T[i][tx] = moe[(size_t)(b * VQ + v0 + i) * SQ + s0 + tx];
    }
    __syncthreads();
#pragma unroll
    for (int r = 0; r < 4; ++r) {
        int i = r * 8 + ty;
        size_t idx = (size_t)(b * SQ + s0 + i) * VQ + v0 + tx;
        x2[idx] = T[tx][i] + x[idx];
    }
}

// ---------------------------------------------------------------------------
// Kernels 4/5: generic 128x128-tile fp32 WMMA GEMM: out = op(A*W^T + bias [+res])
// A: [M,KDIM] row-major, W: [N,KDIM] row-major.
// ---------------------------------------------------------------------------
template <int KDIM, bool RELU, bool RES>
__global__ __launch_bounds__(256) void gemm128(
    const float* __restrict__ A, const float* __restrict__ W,
    const float* __restrict__ bias, const float* __restrict__ res,
    float* __restrict__ out, int N) {
    __shared__ float As[128 * LDK];
    __shared__ float Ws[128 * LDK];

    const int tid = threadIdx.x;
    const int rowBase = blockIdx.x * 128;
    const int colBase = blockIdx.y * 128;
    const int wid = tid >> 5, lane = tid & 31;
    const int lr = lane & 15;
    const int hk = (lane >> 4) * 2;
    const int wr = (wid >> 2) * 64;
    const int wc = (wid & 3) * 32;

    v8f acc[4][2];
#pragma unroll
    for (int mi = 0; mi < 4; ++mi)
#pragma unroll
        for (int ni = 0; ni < 2; ++ni) acc[mi][ni] = v8zero();

    for (int kc = 0; kc < KDIM; kc += 32) {
#pragma unroll
        for (int p = 0; p < 4; ++p) {
            int lin = p * 256 + tid;
            int rl = lin >> 3, s4 = (lin & 7) * 4;
            g2l_b128(&As[rl * LDK + s4],
                     &A[(size_t)(rowBase + rl) * KDIM + kc + s4]);
            g2l_b128(&Ws[rl * LDK + s4],
                     &W[(size_t)(colBase + rl) * KDIM + kc + s4]);
        }
        async_join();
        __syncthreads();
#pragma unroll
        for (int kk = 0; kk < 32; kk += 4) {
            v2f a[4], bb[2];
#pragma unroll
            for (int mi = 0; mi < 4; ++mi) {
                a[mi].x = As[(wr + mi * 16 + lr) * LDK + kk + hk];
                a[mi].y = As[(wr + mi * 16 + lr) * LDK + kk + hk + 1];
            }
#pragma unroll
            for (int ni = 0; ni < 2; ++ni) {
                bb[ni].x = Ws[(wc + ni * 16 + lr) * LDK + kk + hk];
                bb[ni].y = Ws[(wc + ni * 16 + lr) * LDK + kk + hk + 1];
            }
#pragma unroll
            for (int mi = 0; mi < 4; ++mi)
#pragma unroll
                for (int ni = 0; ni < 2; ++ni)
                    acc[mi][ni] = wmma4(a[mi], bb[ni], acc[mi][ni]);
        }
        __syncthreads();
    }
#pragma unroll
    for (int ni = 0; ni < 2; ++ni) {
        int n = colBase + wc + ni * 16 + lr;
        float bv = bias[n];
#pragma unroll
        for (int mi = 0; mi < 4; ++mi) {
#pragma unroll
            for (int i = 0; i < 8; ++i) {
                int m = rowBase + wr + mi * 16 + (lane >> 4) * 8 + i;
                float v = acc[mi][ni][i] + bv;
                if (RES) v += res[(size_t)m * N + n];
                if (RELU) v = fmaxf(v, 0.f);
                out[(size_t)m * N + n] = v;
            }
        }
    }
}

// ---------------------------------------------------------------------------
extern "C" void kernel_launch(void* const* d_in, const int* in_sizes, int n_in,
                              void* d_out, int out_size, void* d_ws,
                              size_t ws_size, hipStream_t stream) {
    const float* x   = (const float*)d_in[0];   // [B,S,V]
    const float* gW  = (const float*)d_in[1];   // [E,S]
    const float* eW  = (const float*)d_in[2];   // [E,S,S]
    const float* eB  = (const float*)d_in[3];   // [E,S]
    const float* f1W = (const float*)d_in[4];   // [FF,V]
    const float* f1b = (const float*)d_in[5];   // [FF]
    const float* f2W = (const float*)d_in[6];   // [V,FF]
    const float* f2b = (const float*)d_in[7];   // [V]

    float* xout  = (float*)d_out;                       // [B,S,V]
    float* probs = xout + (size_t)BQ * SQ * VQ;         // [B,V,E]

    // workspace layout: full_w | x2 | (moe aliased by h; moe dead before h)
    float* fullw = (float*)d_ws;                        // TOK*EQ
    float* x2    = fullw + (size_t)TOK * EQ;            // M2*VQ   (32 MB)
    float* moe   = x2 + (size_t)M2 * VQ;                // TOK*SQ  (32 MB)
    float* h     = moe;                                 // M2*FFQ  (128 MB, reuses moe)

    gate_softmax_topk<<<TOK / 128, 256, 0, stream>>>(x, gW, probs, fullw);

    moe_expert_gemm<<<dim3(TOK / 128, SQ / 128), 256, 0, stream>>>(x, eW, eB,
                                                                   fullw, moe);

    transpose_res<<<dim3(SQ / 32, VQ / 32, BQ), 256, 0, stream>>>(moe, x, x2);

    gemm128<VQ, true, false>
        <<<dim3(M2 / 128, FFQ / 128), 256, 0, stream>>>(x2, f1W, f1b, nullptr,
                                                        h, FFQ);

    gemm128<FFQ, false, true>
        <<<dim3(M2 / 128, VQ / 128), 256, 0, stream>>>(h, f2W, f2b, x2, xout,
                                                       VQ);
}